// GCN_70196945485932
// MI455X (gfx1250) — compile-verified
//
#include <hip/hip_runtime.h>
#include <math.h>

#define N_NODES 20000
#define NMETA   64
#define SLOPE   0.2f

typedef __attribute__((ext_vector_type(2))) float v2f;
typedef __attribute__((ext_vector_type(8))) float v8f;

// ---------------- hypernetwork GEMV: h = meta @ W + b ----------------
__global__ void gemv64_kernel(const float* __restrict__ meta,
                              const float* __restrict__ W,
                              const float* __restrict__ b,
                              float* __restrict__ h, int ncols) {
  int j = blockIdx.x * blockDim.x + threadIdx.x;
  if (j >= ncols) return;
  float acc = b[j];
#pragma unroll 8
  for (int m = 0; m < NMETA; ++m) acc += meta[m] * W[(size_t)m * ncols + j];
  h[j] = acc;
}

// ---------------- layer-1 feature transform (K = 1, trivial) ----------------
__global__ void l1_feat_kernel(const float* __restrict__ X,
                               const float* __restrict__ w,
                               float* __restrict__ feat) {
  int t = blockIdx.x * blockDim.x + threadIdx.x;
  if (t >= N_NODES * 128) return;
  feat[t] = X[t >> 7] * w[t & 127];
}

// ---------------- WMMA fp32 GEMM: feat[M=N][F] = X[M][K] * W(f,k)^T --------
// W row f lives at W[(f>>5)*hS + (f&31)*K + extra + k]  (covers all layer
// layouts; dense case is hS = 32*K, extra = 0).
// One wave per 16x16 output tile; K-loop of V_WMMA_F32_16X16X4_F32.
// A 16x4 fp32 layout: lane<16 -> M=lane, K={0,1}; lane>=16 -> M=lane-16, K={2,3}.
// B 4x16 / C 16x16: row striped across lanes within a VGPR (ISA 7.12.2).
__global__ void wmma_feat_kernel(const float* __restrict__ X,
                                 const float* __restrict__ W,
                                 float* __restrict__ feat,
                                 int K, int F, int hS, int extra) {
  const int lane  = threadIdx.x & 31;
  const int m0    = blockIdx.x << 4;
  const int f0    = blockIdx.y << 4;
  const int mrow  = lane & 15;
  const int kHalf = (lane >> 4) << 1;          // 0 or 2
  const int n     = lane & 15;
  const int f     = f0 + n;

  const float* __restrict__ xrow = X + (size_t)(m0 + mrow) * K;
  const float* __restrict__ wrow = W + (size_t)(f >> 5) * hS + (size_t)(f & 31) * K + extra;

  v8f c = {};
  for (int k0 = 0; k0 < K; k0 += 4) {
    v2f a, b;
    a.x = xrow[k0 + kHalf];
    a.y = xrow[k0 + kHalf + 1];
    b.x = wrow[k0 + kHalf];
    b.y = wrow[k0 + kHalf + 1];
    c = __builtin_amdgcn_wmma_f32_16x16x4_f32(false, a, false, b,
                                              (short)0, c, false, false);
  }
  const int rOff = (lane >> 4) << 3;           // 0 or 8
#pragma unroll
  for (int g = 0; g < 8; ++g)
    feat[(size_t)(m0 + g + rOff) * F + f0 + n] = c[g];
}

// ---------------- per-node attention dots ----------------
// attn_l[f=h*32+d] = A[(h>>1)*aS + (h&1)*32 + d + aX]; attn_r = same + 2*aS.
__global__ void el_er_kernel(const float* __restrict__ feat,
                             const float* __restrict__ A,
                             int aS, int aX,
                             float* __restrict__ el, float* __restrict__ er,
                             int H) {
  int t = blockIdx.x * blockDim.x + threadIdx.x;   // n*H + h
  if (t >= N_NODES * H) return;
  int n = t / H, h = t % H;
  int F = H << 5;
  const float* frow = feat + (size_t)n * F + (h << 5);
  int base = (h >> 1) * aS + ((h & 1) << 5) + aX;
  float sl = 0.f, sr = 0.f;
#pragma unroll
  for (int d = 0; d < 32; ++d) {
    float fv = frow[d];
    sl += fv * A[base + d];
    sr += fv * A[2 * aS + base + d];
  }
  el[t] = sl;
  er[t] = sr;
}

// ---------------- init: m = -inf, s = 0 ----------------
__global__ void init_ms_kernel(float* __restrict__ m, float* __restrict__ s, int nh) {
  int t = blockIdx.x * blockDim.x + threadIdx.x;
  if (t >= nh) return;
  m[t] = -INFINITY;
  s[t] = 0.f;
}

// ---------------- init: out = broadcast bias ----------------
// bias[f] = B[(f>>5)*bhS + (f&31) + bX]
__global__ void init_out_kernel(float* __restrict__ out,
                                const float* __restrict__ B,
                                int bhS, int bX, int F) {
  int t = blockIdx.x * blockDim.x + threadIdx.x;
  if (t >= N_NODES * F) return;
  int fidx = t % F;
  out[t] = B[(fidx >> 5) * bhS + (fidx & 31) + bX];
}

// ---------------- float atomic max via sign-split int trick ----------------
__device__ __forceinline__ void atomicMaxFloat(float* addr, float v) {
  if (v >= 0.f) atomicMax((int*)addr, __float_as_int(v));
  else          atomicMin((unsigned int*)addr, __float_as_uint(v));
}

__device__ __forceinline__ float lrelu(float x) {
  return x > 0.f ? x : SLOPE * x;
}

// ---------------- edge pass 1: segment max of e over dst ----------------
__global__ void edge_max_kernel(const int* __restrict__ src,
                                const int* __restrict__ dst,
                                const float* __restrict__ el,
                                const float* __restrict__ er,
                                float* __restrict__ m, int H, int E) {
  int e = blockIdx.x * blockDim.x + threadIdx.x;
  if (e >= E) return;
  int sn = src[e], dn = dst[e];
  for (int h = 0; h < H; ++h) {
    float v = lrelu(el[sn * H + h] + er[dn * H + h]);
    atomicMaxFloat(&m[dn * H + h], v);
  }
}

// ---------------- edge pass 2: ex = exp(e - m[dst]); s += ex ----------------
__global__ void edge_exp_kernel(const int* __restrict__ src,
                                const int* __restrict__ dst,
                                const float* __restrict__ el,
                                const float* __restrict__ er,
                                const float* __restrict__ m,
                                float* __restrict__ s,
                                float* __restrict__ exb, int H, int E) {
  int e = blockIdx.x * blockDim.x + threadIdx.x;
  if (e >= E) return;
  int sn = src[e], dn = dst[e];
  for (int h = 0; h < H; ++h) {
    float v  = lrelu(el[sn * H + h] + er[dn * H + h]);
    float ex = __expf(v - m[dn * H + h]);
    exb[(size_t)e * H + h] = ex;
    atomicAdd(&s[dn * H + h], ex);
  }
}

// ---------------- edge pass 3: out[dst] += alpha * feat[src] ----------------
// One wave32 per edge; lane = feature lane within head -> coalesced atomics.
__global__ void edge_aggr_kernel(const int* __restrict__ src,
                                 const int* __restrict__ dst,
                                 const float* __restrict__ feat,
                                 const float* __restrict__ exb,
                                 const float* __restrict__ s,
                                 float* __restrict__ out, int H, int E) {
  int wid  = (blockIdx.x * blockDim.x + threadIdx.x) >> 5;   // edge id
  int lane = threadIdx.x & 31;
  if (wid >= E) return;
  int sn = src[wid], dn = dst[wid];
  int F = H << 5;
  for (int h = 0; h < H; ++h) {
    float alpha = exb[(size_t)wid * H + h] / (s[dn * H + h] + 1e-9f);
    float v = alpha * feat[(size_t)sn * F + (h << 5) + lane];
    atomicAdd(&out[(size_t)dn * F + (h << 5) + lane], v);
  }
}

// ---------------- elementwise ELU ----------------
__global__ void elu_kernel(float* __restrict__ x, int n) {
  int t = blockIdx.x * blockDim.x + threadIdx.x;
  if (t >= n) return;
  float v = x[t];
  x[t] = v > 0.f ? v : (__expf(v) - 1.f);
}

// ---------------- host-side layer driver ----------------
static void gat_after_feat(hipStream_t stream, int H,
                           const float* A, int aS, int aX,
                           const float* B, int bhS, int bX,
                           const int* src, const int* dst, int E,
                           const float* feat, float* el, float* er,
                           float* m, float* s, float* exb, float* out) {
  int F  = H << 5;
  int NH = N_NODES * H;
  el_er_kernel<<<(NH + 255) / 256, 256, 0, stream>>>(feat, A, aS, aX, el, er, H);
  init_ms_kernel<<<(NH + 255) / 256, 256, 0, stream>>>(m, s, NH);
  init_out_kernel<<<(N_NODES * F + 255) / 256, 256, 0, stream>>>(out, B, bhS, bX, F);
  edge_max_kernel<<<(E + 255) / 256, 256, 0, stream>>>(src, dst, el, er, m, H, E);
  edge_exp_kernel<<<(E + 255) / 256, 256, 0, stream>>>(src, dst, el, er, m, s, exb, H, E);
  edge_aggr_kernel<<<((size_t)E * 32 + 255) / 256, 256, 0, stream>>>(src, dst, feat, exb, s, out, H, E);
}

extern "C" void kernel_launch(void* const* d_in, const int* in_sizes, int n_in,
                              void* d_out, int out_size, void* d_ws, size_t ws_size,
                              hipStream_t stream) {
  const float* X      = (const float*)d_in[0];
  const float* meta1  = (const float*)d_in[1];
  const float* meta2  = (const float*)d_in[2];
  const float* Whin   = (const float*)d_in[3];
  const float* bhin   = (const float*)d_in[4];
  const float* Whout  = (const float*)d_in[5];
  const float* bhout  = (const float*)d_in[6];
  const int*   src_in = (const int*)d_in[7];
  const int*   dst_in = (const int*)d_in[8];
  const int*   src_tg = (const int*)d_in[9];
  const int*   dst_tg = (const int*)d_in[10];
  const int    E      = in_sizes[7];

  // workspace carve-up (floats); total ~8.7M floats (~35 MB)
  float* ws   = (float*)d_ws;
  float* h1   = ws;                               // 4704
  float* h2   = h1 + 4704;                        // 26016
  float* feat = h2 + 26016;                       // N*128
  float* el   = feat + (size_t)N_NODES * 128;     // N*4
  float* er   = el + (size_t)N_NODES * 4;         // N*4
  float* m    = er + (size_t)N_NODES * 4;         // N*4
  float* s    = m + (size_t)N_NODES * 4;          // N*4
  float* exb  = s + (size_t)N_NODES * 4;          // E*4
  float* bufA = exb + (size_t)E * 4;              // N*128
  float* bufB = bufA + (size_t)N_NODES * 128;     // N*32
  float* out  = (float*)d_out;                    // N*32

  // hypernetwork parameter vectors
  gemv64_kernel<<<(4704 + 255) / 256, 256, 0, stream>>>(meta1, Whin, bhin, h1, 4704);
  gemv64_kernel<<<(26016 + 255) / 256, 256, 0, stream>>>(meta2, Whout, bhout, h2, 26016);

  // h1 layout: a_f@0(256) w_f@256(128) b_f@384(128) ao@512(64) wo@576(4096) bo@4672(32)
  // h2 layout: a@0(768) w@768(12288) b@13056(384) ao@13440(192) wo@13632(12288) bo@25920(96)

  dim3 blk32(32);

  // --- Layer 1: GAT(x(N,1) -> H=4, dout=32) on (src_in,dst_in) -> bufA ---
  l1_feat_kernel<<<(N_NODES * 128 + 255) / 256, 256, 0, stream>>>(X, h1 + 256, feat);
  gat_after_feat(stream, 4, h1 + 0, 64, 0, h1 + 384, 32, 0,
                 src_in, dst_in, E, feat, el, er, m, s, exb, bufA);

  // --- Layer 2: GAT(bufA(N,128) -> H=1, dout=32) -> bufB, then ELU ---
  wmma_feat_kernel<<<dim3(N_NODES / 16, 2), blk32, 0, stream>>>(
      bufA, h1 + 576, feat, 128, 32, 32 * 128, 0);
  gat_after_feat(stream, 1, h1 + 512, 16, 0, h1 + 4672, 32, 0,
                 src_in, dst_in, E, feat, el, er, m, s, exb, bufB);
  elu_kernel<<<(N_NODES * 32 + 255) / 256, 256, 0, stream>>>(bufB, N_NODES * 32);

  // --- Layer 3: GAT(bufB(N,32) -> H=4) on (src_trg,dst_trg) -> bufA ---
  // W row f at (f>>5)*3072 + (f&31)*32 ; bias (f>>5)*96 + (f&31) ; attn aS=192
  wmma_feat_kernel<<<dim3(N_NODES / 16, 8), blk32, 0, stream>>>(
      bufB, h2 + 768, feat, 32, 128, 3072, 0);
  gat_after_feat(stream, 4, h2 + 0, 192, 0, h2 + 13056, 96, 0,
                 src_tg, dst_tg, E, feat, el, er, m, s, exb, bufA);

  // --- Layer 4: GAT(bufA(N,128) -> H=1) -> bufB, ELU ---
  wmma_feat_kernel<<<dim3(N_NODES / 16, 2), blk32, 0, stream>>>(
      bufA, h2 + 13632, feat, 128, 32, 32 * 128, 0);
  gat_after_feat(stream, 1, h2 + 13440, 16, 0, h2 + 25920, 32, 0,
                 src_tg, dst_tg, E, feat, el, er, m, s, exb, bufB);
  elu_kernel<<<(N_NODES * 32 + 255) / 256, 256, 0, stream>>>(bufB, N_NODES * 32);

  // --- Layer 5: GAT(bufB(N,32) -> H=4) -> bufA  (second slice of h2 params) ---
  wmma_feat_kernel<<<dim3(N_NODES / 16, 8), blk32, 0, stream>>>(
      bufB, h2 + 768, feat, 32, 128, 3072, 1024);
  gat_after_feat(stream, 4, h2 + 0, 192, 64, h2 + 13056, 96, 32,
                 src_tg, dst_tg, E, feat, el, er, m, s, exb, bufA);

  // --- Layer 6: GAT(bufA(N,128) -> H=1) -> d_out, ELU in place ---
  wmma_feat_kernel<<<dim3(N_NODES / 16, 2), blk32, 0, stream>>>(
      bufA, h2 + 13632 + 4096, feat, 128, 32, 32 * 128, 0);
  gat_after_feat(stream, 1, h2 + 13504, 16, 0, h2 + 25952, 32, 0,
                 src_tg, dst_tg, E, feat, el, er, m, s, exb, out);
  elu_kernel<<<(N_NODES * 32 + 255) / 256, 256, 0, stream>>>(out, N_NODES * 32);
}